// SplatTextureModule_28664611733893
// MI455X (gfx1250) — compile-verified
//
#include <hip/hip_runtime.h>

#define C_FEAT 32
#define CH_OUT 33

typedef float __attribute__((ext_vector_type(2))) float2v;
typedef float __attribute__((ext_vector_type(4))) float4v;

// ---------------------------------------------------------------------------
// Bilinear scatter splat. One wave32 per 32-point batch.
//   Phase 1: lane p computes corner offsets/weights for point (base+p)
//            (coalesced NT float2 uv load).
//   Phase 2: per point i (uniform loop index): v_readlane broadcasts the 4
//            corner offsets + 4 weights to SGPRs; lane c handles channel c ->
//            4 fully-packed global_atomic_add_f32 per point (all 32 lanes
//            active, scalar base + constant lane offset addressing). The
//            channel-row load for point i+1 is issued before point i's
//            atomics to pipeline HBM latency.
//   Phase 3: homogeneous weight channel (ch 32) batched across lanes:
//            lane p issues 4 packed atomics for its own point.
// Inputs are streamed with non-temporal hints so the 138 MB texture stays
// resident in the 192 MB L2 and all 264M atomics resolve at the L2 atomic
// units.
// (Defined first in the file so its disassembly appears in the feedback
//  snippet for verification of fadd-atomic + TH_NT codegen.)
// ---------------------------------------------------------------------------
__global__ void splat_scatter_kernel(const float* __restrict__ values,
                                     const float* __restrict__ uv,
                                     const int*  __restrict__ tsize_ptr,
                                     float* __restrict__ tex,
                                     int nPoints) {
    const int lane          = threadIdx.x & 31;
    const int wavesPerBlock = blockDim.x >> 5;
    const int waveId        = blockIdx.x * wavesPerBlock + (threadIdx.x >> 5);
    const int totalWaves    = gridDim.x * wavesPerBlock;

    const int   T   = *tsize_ptr;       // uniform scalar load
    const float Tm1 = (float)(T - 1);
    const int nBatches = (nPoints + 31) >> 5;

    for (int batch = waveId; batch < nBatches; batch += totalWaves) {
        const int base = batch << 5;
        const int pt   = base + lane;
        const int npts = min(32, nPoints - base);

        // ---- Phase 1: per-lane corner offsets + weights for its point ----
        int   s0 = 0, s1 = 0, s2 = 0, s3 = 0;
        float w0 = 0.f, w1 = 0.f, w2 = 0.f, w3 = 0.f;
        if (pt < nPoints) {
            float2v uvp = __builtin_nontemporal_load(
                (const float2v*)(uv + 2l * (long)pt));
            float x  = uvp.x * Tm1;
            float y  = uvp.y * Tm1;
            float xf = floorf(x), yf = floorf(y);
            float wx1 = x - xf, wy1 = y - yf;
            float wx0 = 1.f - wx1, wy0 = 1.f - wy1;
            int x0 = (int)xf;  x0 = max(0, min(T - 1, x0));
            int y0 = (int)yf;  y0 = max(0, min(T - 1, y0));
            int x1 = min(T - 1, x0 + 1);
            int y1 = min(T - 1, y0 + 1);
            s0 = (y0 * T + x0) * CH_OUT;
            s1 = (y0 * T + x1) * CH_OUT;
            s2 = (y1 * T + x0) * CH_OUT;
            s3 = (y1 * T + x1) * CH_OUT;
            w0 = wy0 * wx0;  w1 = wy0 * wx1;
            w2 = wy1 * wx0;  w3 = wy1 * wx1;
        }

        // ---- Phase 2: channel-parallel atomics, one point per iteration ----
        const float* __restrict__ vrow = values + (long)base * C_FEAT;
        float v = (npts > 0) ? __builtin_nontemporal_load(vrow + lane) : 0.f;
        for (int i = 0; i < npts; ++i) {
            // prefetch next point's channel row (hides load latency behind atomics)
            float vn = (i + 1 < npts)
                ? __builtin_nontemporal_load(vrow + (long)(i + 1) * C_FEAT + lane)
                : 0.f;
            // uniform broadcasts of point i's scalars (v_readlane -> SGPR)
            int   b0 = __builtin_amdgcn_readlane(s0, i);
            int   b1 = __builtin_amdgcn_readlane(s1, i);
            int   b2 = __builtin_amdgcn_readlane(s2, i);
            int   b3 = __builtin_amdgcn_readlane(s3, i);
            float u0 = __int_as_float(__builtin_amdgcn_readlane(__float_as_int(w0), i));
            float u1 = __int_as_float(__builtin_amdgcn_readlane(__float_as_int(w1), i));
            float u2 = __int_as_float(__builtin_amdgcn_readlane(__float_as_int(w2), i));
            float u3 = __int_as_float(__builtin_amdgcn_readlane(__float_as_int(w3), i));
            // 4 atomics, all 32 lanes active, lane c -> channel c
            (void)unsafeAtomicAdd(tex + b0 + lane, u0 * v);
            (void)unsafeAtomicAdd(tex + b1 + lane, u1 * v);
            (void)unsafeAtomicAdd(tex + b2 + lane, u2 * v);
            (void)unsafeAtomicAdd(tex + b3 + lane, u3 * v);
            v = vn;
        }

        // ---- Phase 3: homogeneous weight channel (ch 32), lane p = point p ----
        if (pt < nPoints) {
            (void)unsafeAtomicAdd(tex + s0 + C_FEAT, w0);
            (void)unsafeAtomicAdd(tex + s1 + C_FEAT, w1);
            (void)unsafeAtomicAdd(tex + s2 + C_FEAT, w2);
            (void)unsafeAtomicAdd(tex + s3 + C_FEAT, w3);
        }
    }
}

// ---------------------------------------------------------------------------
// Zero-fill the output texture (harness poisons d_out with 0xAA).
// Plain RT stores on purpose: we WANT the zeroed texture resident in L2 so
// the subsequent 264M atomics hit the L2 atomic units.
// ---------------------------------------------------------------------------
__global__ void splat_zero_kernel(float* __restrict__ out, long n) {
    const long n4     = n >> 2;
    const long stride = (long)gridDim.x * blockDim.x;
    long i            = (long)blockIdx.x * blockDim.x + threadIdx.x;
    float4v z = {0.f, 0.f, 0.f, 0.f};
    for (; i < n4; i += stride) {
        ((float4v* __restrict__)out)[i] = z;
    }
    // scalar tail (generic; no-op when n % 4 == 0)
    const long tail = n & 3;
    if (blockIdx.x == 0 && (long)threadIdx.x < tail) {
        out[(n4 << 2) + threadIdx.x] = 0.f;
    }
}

extern "C" void kernel_launch(void* const* d_in, const int* in_sizes, int n_in,
                              void* d_out, int out_size, void* d_ws, size_t ws_size,
                              hipStream_t stream) {
    const float* values = (const float*)d_in[0];
    const float* uv     = (const float*)d_in[1];
    const int*   tsize  = (const int*)d_in[2];
    float*       out    = (float*)d_out;

    const int nPoints = in_sizes[0] / C_FEAT;

    // 1) zero the texture (stream order guarantees completion before scatter)
    {
        long n = (long)out_size;
        int  blocks = 4096;
        splat_zero_kernel<<<blocks, 256, 0, stream>>>(out, n);
    }

    // 2) scatter splat: one wave32 per 32-point batch
    {
        const int nBatches      = (nPoints + 31) / 32;
        const int wavesPerBlock = 8;   // 256 threads
        const int blocks        = (nBatches + wavesPerBlock - 1) / wavesPerBlock;
        splat_scatter_kernel<<<blocks, 256, 0, stream>>>(values, uv, tsize, out, nPoints);
    }
}